// GSpectralNet_13065290514687
// MI455X (gfx1250) — compile-verified
//
#include <hip/hip_runtime.h>

typedef __attribute__((ext_vector_type(2))) float v2f;
typedef __attribute__((ext_vector_type(8))) float v8f;

#define FDIM 128
#define PSTRIDE 288   // DWORDs per K-pair row in LDS (256 data + 32 pad for bank rotation)

// ---------------------------------------------------------------- degree prep
__global__ void __launch_bounds__(256) k_fill_one(float* __restrict__ p, int n) {
    int i = blockIdx.x * blockDim.x + threadIdx.x;
    if (i < n) p[i] = 1.0f;
}

__global__ void __launch_bounds__(256) k_deg(const int* __restrict__ eidx,
                                             float* __restrict__ deg, int e) {
    int i = blockIdx.x * blockDim.x + threadIdx.x;
    if (i < e) atomicAdd(&deg[eidx[e + i]], 1.0f);   // dst = eidx[E + i]
}

__global__ void __launch_bounds__(256) k_rsqrt_inplace(float* __restrict__ p, int n) {
    int i = blockIdx.x * blockDim.x + threadIdx.x;
    if (i < n) p[i] = rsqrtf(p[i]);
}

// ---------------------------------------------------------------- WMMA GEMM
// Y[nrows x 128] = X[nrows x 128] @ W[128 x 128] + b, fp32 via v_wmma_f32_16x16x4_f32.
// W staged in LDS in K-pair-swizzled form: element (k, col) lives at
//   sW[(k>>1)*PSTRIDE + col*2 + (k&1)]
// so a B fragment {K, K+1} x col is one aligned 8-byte ds_load_b64 into an even
// VGPR pair (what WMMA SRC1 wants), and the 288-DWORD pair-row stride rotates
// banks so lanes 0-15 (pair p) and lanes 16-31 (pair p+1) use disjoint bank halves.
__global__ void __launch_bounds__(256) k_gemm_bias(const float* __restrict__ X,
                                                   const float* __restrict__ W,
                                                   const float* __restrict__ b,
                                                   float* __restrict__ Y,
                                                   int nrows) {
    __shared__ float sW[64 * PSTRIDE];
    for (int i = threadIdx.x; i < FDIM * FDIM; i += 256) {
        int k   = i >> 7;
        int col = i & 127;
        sW[(k >> 1) * PSTRIDE + (col << 1) + (k & 1)] = W[i];
    }
    __syncthreads();

    const int lane    = threadIdx.x & 31;
    const int wave    = threadIdx.x >> 5;
    const int m0      = (blockIdx.x * 8 + wave) * 16;
    if (m0 >= nrows) return;                     // wave-uniform: EXEC stays all-1s

    const int halfsel = lane >> 4;               // 0 = lanes 0-15, 1 = lanes 16-31
    const int lm      = lane & 15;

    v8f c[8];
    #pragma unroll
    for (int t = 0; t < 8; ++t) {
        float bv = b[t * 16 + lm];               // bias depends only on column
        #pragma unroll
        for (int r = 0; r < 8; ++r) c[t][r] = bv;
    }

    // A fragment (ISA 16x4 f32 layout): lane row = m0+lm, K pair = kb + 2*halfsel
    const float* arow = X + (size_t)(m0 + lm) * FDIM + (halfsel << 1);

    for (int kb = 0; kb < FDIM; kb += 4) {
        v2f a = *(const v2f*)(arow + kb);        // {K, K+1} for this lane
        // B fragments: pair row p = kb/2 + halfsel, column = t*16 + lm
        const float* bbase = &sW[((kb >> 1) + halfsel) * PSTRIDE + (lm << 1)];
        #pragma unroll
        for (int t = 0; t < 8; ++t) {
            v2f bb = *(const v2f*)(bbase + t * 32);   // one ds_load_b64
            c[t] = __builtin_amdgcn_wmma_f32_16x16x4_f32(
                false, a, false, bb, (short)0, c[t], false, false);
        }
    }

    // C/D layout: VGPR r -> M = r + 8*halfsel, N = lm
    #pragma unroll
    for (int t = 0; t < 8; ++t) {
        float* yp = Y + (size_t)(m0 + (halfsel << 3)) * FDIM + t * 16 + lm;
        #pragma unroll
        for (int r = 0; r < 8; ++r) yp[(size_t)r * FDIM] = c[t][r];
    }
}

// ---------------------------------------------------------------- self-loop init
// out[i,:] = dinv[i]^2 * Y[i,:]   (fully overwrites d_out slab before scatter)
__global__ void __launch_bounds__(256) k_selfloop(const float* __restrict__ Y,
                                                  const float* __restrict__ dinv,
                                                  float* __restrict__ out, int n) {
    int i = blockIdx.x * blockDim.x + threadIdx.x;   // over n*32 float4 slots
    if (i >= n * 32) return;
    int row = i >> 5;
    float d = dinv[row];
    d = d * d;
    float4 v = ((const float4*)Y)[i];
    float4 o;
    o.x = v.x * d; o.y = v.y * d; o.z = v.z * d; o.w = v.w * d;
    ((float4*)out)[i] = o;
}

// ---------------------------------------------------------------- edge scatter
// One wave per edge: 32 lanes x float4 = one 128-float feature row, coalesced
// 512B gather of Y[src,:], then 128 non-returning f32 atomics into out[dst,:].
__global__ void __launch_bounds__(256) k_scatter(const int* __restrict__ eidx,
                                                 const float* __restrict__ Y,
                                                 const float* __restrict__ dinv,
                                                 float* __restrict__ out, int e) {
    int edge = blockIdx.x * 8 + (threadIdx.x >> 5);
    if (edge >= e) return;
    int lane = threadIdx.x & 31;
    int src = eidx[edge];
    int dst = eidx[e + edge];
    float nrm = dinv[src] * dinv[dst];
    float4 v = *(const float4*)(Y + (size_t)src * FDIM + lane * 4);
    float* od = out + (size_t)dst * FDIM + lane * 4;
    atomicAdd(od + 0, v.x * nrm);
    atomicAdd(od + 1, v.y * nrm);
    atomicAdd(od + 2, v.z * nrm);
    atomicAdd(od + 3, v.w * nrm);
}

// ---------------------------------------------------------------- launcher
extern "C" void kernel_launch(void* const* d_in, const int* in_sizes, int n_in,
                              void* d_out, int out_size, void* d_ws, size_t ws_size,
                              hipStream_t stream) {
    const float* X[3]   = {(const float*)d_in[0], (const float*)d_in[1], (const float*)d_in[2]};
    const int*   EI[3]  = {(const int*)d_in[3],   (const int*)d_in[4],   (const int*)d_in[5]};
    const float* W[3]   = {(const float*)d_in[6], (const float*)d_in[8], (const float*)d_in[10]};
    const float* B[3]   = {(const float*)d_in[7], (const float*)d_in[9], (const float*)d_in[11]};

    const int n = in_sizes[0] / FDIM;   // 50000
    const int e = in_sizes[3] / 2;      // 800000
    const size_t nf = (size_t)n * FDIM;

    float* ws = (float*)d_ws;           // layout: [3*nf] Y  |  [3*n] dinv

    for (int m = 0; m < 3; ++m) {
        float* Y    = ws + (size_t)m * nf;
        float* dinv = ws + 3 * nf + (size_t)m * n;
        float* out  = (float*)d_out + (size_t)m * nf;

        k_fill_one<<<(n + 255) / 256, 256, 0, stream>>>(dinv, n);
        k_deg<<<(e + 255) / 256, 256, 0, stream>>>(EI[m], dinv, e);
        k_rsqrt_inplace<<<(n + 255) / 256, 256, 0, stream>>>(dinv, n);

        int rowTiles = (n + 15) / 16;                       // 3125
        k_gemm_bias<<<(rowTiles + 7) / 8, 256, 0, stream>>>(X[m], W[m], B[m], Y, n);

        k_selfloop<<<(n * 32 + 255) / 256, 256, 0, stream>>>(Y, dinv, out, n);
        k_scatter<<<(e + 7) / 8, 256, 0, stream>>>(EI[m], Y, dinv, out, e);
    }
}